// CategoricalGumbelSoftmax_5531917877942
// MI455X (gfx1250) — compile-verified
//
#include <hip/hip_runtime.h>
#include <hip/hip_bf16.h>

typedef __attribute__((ext_vector_type(2))) float v2f;
typedef __attribute__((ext_vector_type(8))) float v8f;

#define GB 2      // batch
#define GG 512    // genes
#define GD 256    // embed dim
#define NC 3      // categories

// ---------------------------------------------------------------------------
// Kernel 1: hsb = GE @ W1[:, :D]^T + b1  and  htb = GE @ W1[:, D:]^T
// One wave (32 threads) per 16x16 output tile, exact-f32 WMMA 16x16x4.
// A[m,k] = GE[m0+m, k]      (row-major, contiguous in k -> b64 loads)
// B[k,n] = W1[n0+n, sel*D+k](row-major, contiguous in k -> b64 loads)
// A/B lane layout: lanes 0-15 hold K={0,1}, lanes 16-31 hold K={2,3}.
// ---------------------------------------------------------------------------
__global__ __launch_bounds__(32) void gumbel_hs_ht_gemm(
    const float* __restrict__ ge, const float* __restrict__ W1,
    const float* __restrict__ b1, float* __restrict__ hsb,
    float* __restrict__ htb)
{
    const int lane = threadIdx.x & 31;
    const int half = lane >> 4;     // which K pair this lane holds
    const int i16  = lane & 15;     // M (for A) / N (for B) index
    const int m0   = blockIdx.x * 16;
    const int n0   = blockIdx.y * 16;
    const int sel  = blockIdx.z;    // 0 -> hs (+b1), 1 -> ht

    const float* Arow = ge + (size_t)(m0 + i16) * GD;
    const float* Brow = W1 + (size_t)(n0 + i16) * (2 * GD) + sel * GD;

    v8f c = {};
#pragma unroll 4
    for (int k0 = 0; k0 < GD; k0 += 4) {
        const int k = k0 + half * 2;
        v2f a  = *(const v2f*)(Arow + k);
        v2f bb = *(const v2f*)(Brow + k);
        c = __builtin_amdgcn_wmma_f32_16x16x4_f32(
                false, a, false, bb, (short)0, c, false, false);
    }

    const float bias = (sel == 0) ? b1[n0 + i16] : 0.0f;
    float* out = (sel == 0) ? hsb : htb;
    // D layout: lane holds column n=i16; VGPR r holds row m = r + 8*half.
#pragma unroll
    for (int r = 0; r < 8; ++r) {
        const int m = r + half * 8;
        out[(size_t)(m0 + m) * GD + (n0 + i16)] = c[r] + bias;
    }
}

// ---------------------------------------------------------------------------
// Kernel 2: per wave, one (b, s, 16-wide t-tile); 8 waves/block share (b,s).
//   logits[t, c] = sum_h relu(hsb[s,h] + htb[t,h]) * W2[c,h]
// WMMA f32 16x16x4: M = 16 t's, N = 16 (c<3 live), K = h in steps of 4.
// LDS staging: hs row (shared by whole block) and W2 as a 4-row table with
// row 3 zeroed -> hot loop is 1 global b64 (ht) + 2 ds b64 + relu + wmma,
// no per-iteration selects. Epilogue: regroup 3 logits/pair via LDS, add
// b2 + gumbel(-log(-log u)), first-max argmax (matches jnp.argmax), emit
// one-hot y (forward value of straight-through y is exactly y_hard) and M.
// ---------------------------------------------------------------------------
__global__ __launch_bounds__(256) void gumbel_pair_kernel(
    const float* __restrict__ hsb, const float* __restrict__ htb,
    const float* __restrict__ W2, const float* __restrict__ b2,
    const float* __restrict__ u, float* __restrict__ outM,
    float* __restrict__ outY)
{
    __shared__ float s_hs[GD];         // hs row for this block's s (1 KB)
    __shared__ float s_w2[4][GD];      // W2 rows 0..2, row 3 = zeros (4 KB)
    __shared__ float sm[8][16][4];     // [wave][pair][cat(+pad)] (2 KB)

    const int tid  = threadIdx.x;
    const int lane = tid & 31;
    const int w    = tid >> 5;
    const int task = blockIdx.x * 8 + w;    // 32768 tasks total
    const int b    = task >> 14;            // / (512*32)
    const int rem  = task & 16383;
    const int s    = rem >> 5;              // same for all 8 waves in block
    const int t0   = (rem & 31) << 4;

    // ---- stage hs row and padded W2 into LDS (cooperative, 256 threads) ----
    s_hs[tid] = hsb[(size_t)(b * GG + s) * GD + tid];
#pragma unroll
    for (int j = 0; j < 4; ++j) {
        const int idx = tid + j * 256;      // flat into s_w2[4][GD]
        (&s_w2[0][0])[idx] = (idx < NC * GD) ? W2[idx] : 0.0f;
    }
    __syncthreads();

    const int half = lane >> 4;
    const int i16  = lane & 15;
    const float* htrow  = htb + (size_t)(b * GG + t0 + i16) * GD;
    const float* w2row  = s_w2[(i16 < NC) ? i16 : 3];   // row 3 is all zeros

    v8f acc = {};
#pragma unroll 4
    for (int h0 = 0; h0 < GD; h0 += 4) {
        const int k = h0 + half * 2;
        v2f htv = *(const v2f*)(htrow + k);      // global b64 (only stream)
        v2f hsv = *(const v2f*)(s_hs + k);       // ds b64 (broadcast)
        v2f a;
        a.x = fmaxf(hsv.x + htv.x, 0.0f);        // relu(hs + ht), b1 folded
        a.y = fmaxf(hsv.y + htv.y, 0.0f);
        v2f bb = *(const v2f*)(w2row + k);       // ds b64, zero row pads n>=3
        acc = __builtin_amdgcn_wmma_f32_16x16x4_f32(
                  false, a, false, bb, (short)0, acc, false, false);
    }

    // Scatter logits: lane owns category i16 for pairs m = r + 8*half.
    if (i16 < NC) {
#pragma unroll
        for (int r = 0; r < 8; ++r)
            sm[w][r + half * 8][i16] = acc[r];
    }
    __syncthreads();

    if (lane < 16) {
        const int t = t0 + lane;
        const size_t pair = ((size_t)(b * GG + s)) * GG + t;
        const float* up = u + pair * NC;
        float z[NC];
#pragma unroll
        for (int c = 0; c < NC; ++c) {
            const float g = -logf(-logf(up[c]));       // gumbel noise
            z[c] = sm[w][lane][c] + b2[c] + g;         // TAU == 1
        }
        int idx = 0;
        float best = z[0];
        if (z[1] > best) { best = z[1]; idx = 1; }
        if (z[2] > best) { best = z[2]; idx = 2; }

        outM[pair] = (idx == 2) ? 1.0f : ((idx == 0) ? -1.0f : 0.0f);
        float3 y;
        y.x = (idx == 0) ? 1.0f : 0.0f;
        y.y = (idx == 1) ? 1.0f : 0.0f;
        y.z = (idx == 2) ? 1.0f : 0.0f;
        *(float3*)(outY + pair * 3) = y;               // single b96 store
    }
}

// ---------------------------------------------------------------------------
extern "C" void kernel_launch(void* const* d_in, const int* in_sizes, int n_in,
                              void* d_out, int out_size, void* d_ws,
                              size_t ws_size, hipStream_t stream)
{
    const float* ge = (const float*)d_in[0];  // (2,512,256)
    const float* u  = (const float*)d_in[1];  // (2,512,512,3)
    const float* W1 = (const float*)d_in[2];  // (256,512)
    const float* b1 = (const float*)d_in[3];  // (256,)
    const float* W2 = (const float*)d_in[4];  // (3,256)
    const float* b2 = (const float*)d_in[5];  // (3,)

    float* hsb = (float*)d_ws;                         // 1 MB
    float* htb = hsb + (size_t)GB * GG * GD;           // 1 MB
    float* outM = (float*)d_out;                       // (2,512,512)
    float* outY = outM + (size_t)GB * GG * GG;         // (2,512,512,3)

    dim3 g1(GB * GG / 16, GD / 16, 2);                 // 64 x 16 x 2 tiles
    gumbel_hs_ht_gemm<<<g1, 32, 0, stream>>>(ge, W1, b1, hsb, htb);

    dim3 g2(GB * GG * (GG / 16) / 8);                  // 4096 blocks x 8 waves
    gumbel_pair_kernel<<<g2, 256, 0, stream>>>(hsb, htb, W2, b2, u,
                                               outM, outY);
}